// EmbeddingLayer_88450556494723
// MI455X (gfx1250) — compile-verified
//
#include <hip/hip_runtime.h>
#include <hip/hip_bf16.h>

// ---- problem constants (from reference) ----
#define BB 65536
#define KK 1024
#define DD 256
#define SOMW 32
#define LRATE 0.05f

typedef __attribute__((ext_vector_type(2))) float v2f;
typedef __attribute__((ext_vector_type(8))) float v8f;

#define TILE_ROWS 128            // x rows per workgroup (8 waves x 16 rows)
#define XPAD 260                 // padded LDS row stride (floats); 260%64==4 -> conflict-free
#define EBUF (16 * XPAD)         // floats per E-tile buffer
#define NTILE (KK / 16)          // 64 E tiles

// ---------------------------------------------------------------------------
// Kernel 1: zero the delta accumulator and the two loss scalars
// ---------------------------------------------------------------------------
__global__ __launch_bounds__(256) void som_init(float* __restrict__ delta,
                                                float* __restrict__ scalars) {
    int i = blockIdx.x * blockDim.x + threadIdx.x;
    if (i < KK * DD) delta[i] = 0.0f;
    if (i < 2) scalars[i] = 0.0f;
}

// ---------------------------------------------------------------------------
// Kernel 2: per-embedding squared norms (one wave per embedding row)
// ---------------------------------------------------------------------------
__global__ __launch_bounds__(256) void som_enorm(const float* __restrict__ E,
                                                 float* __restrict__ enorm) {
    int w = (blockIdx.x * blockDim.x + threadIdx.x) >> 5;
    int lane = threadIdx.x & 31;
    if (w >= KK) return;
    const float* row = E + w * DD;
    float s = 0.0f;
    #pragma unroll
    for (int j = 0; j < DD / 32; ++j) {
        float v = row[lane + j * 32];
        s += v * v;
    }
    #pragma unroll
    for (int m = 16; m >= 1; m >>= 1) s += __shfl_xor(s, m, 32);
    if (lane == 0) enorm[w] = s;
}

// ---------------------------------------------------------------------------
// Kernel 3: fused distance + argmin via V_WMMA_F32_16X16X4_F32
//   - A operand (16 x-rows per wave) held entirely in VGPRs (64 x v2f),
//     loaded once from global: halves LDS traffic vs LDS-staged A.
//   - E tiles (16 rows) double-buffered in LDS via global_load_async_to_lds,
//     prefetch of tile t+2 overlaps the 64-WMMA chain of tile t+1.
//   score(m,n) = ||e_n||^2 - 2 * dot(x_m, e_n)   (||x_m||^2 is row-constant)
// ---------------------------------------------------------------------------
__global__ __launch_bounds__(256) void som_argmin(const float* __restrict__ X,
                                                  const float* __restrict__ E,
                                                  const float* __restrict__ enorm,
                                                  int* __restrict__ nmin) {
    __shared__ float es[2 * EBUF];            // 2 x 16 x XPAD floats = 33,280 B

    const int tid = threadIdx.x;
    const int lane = tid & 31;
    const int wave = tid >> 5;
    const int rowBase = blockIdx.x * TILE_ROWS;

    const unsigned esbase = (unsigned)(uintptr_t)(&es[0]);   // LDS byte offset

    // async prefetch of E tile t into buffer buf (4 x B128 chunks per thread)
    auto prefetch = [&](int t, int buf) {
        #pragma unroll
        for (int k = 0; k < 4; ++k) {
            int c = tid + k * 256;            // float4 chunk id, 0..1023
            int r = c >> 6;                   // tile row 0..15
            int c4 = c & 63;                  // float4 column
            unsigned voff  = (unsigned)((t * 16 + r) * (DD * 4) + c4 * 16);
            unsigned laddr = esbase + (unsigned)(buf * EBUF + r * XPAD + c4 * 4) * 4u;
            asm volatile("global_load_async_to_lds_b128 %0, %1, %2"
                         :: "v"(laddr), "v"(voff), "s"(E)
                         : "memory");
        }
    };

    // kick off the first two tiles before anything else
    prefetch(0, 0);
    prefetch(1, 1);

    // --- A operand in registers: 16x4 fragments for all 64 k-steps ---
    // lanes 0-15 -> M=lane, K pair {0,1}; lanes 16-31 -> M=lane-16, K pair {2,3}
    const float* xg = X + (size_t)(rowBase + wave * 16 + (lane & 15)) * DD
                        + ((lane >> 4) << 1);
    v2f areg[DD / 4];
    #pragma unroll
    for (int ks = 0; ks < DD / 4; ++ks)
        areg[ks] = *(const v2f*)(xg + ks * 4);

    float rmin[8];
    int   ridx[8];
    #pragma unroll
    for (int i = 0; i < 8; ++i) { rmin[i] = 3.4e38f; ridx[i] = 0x7fffffff; }

    // B fragment offset inside an E buffer: row n = lane&15, same K-pair pattern
    const int eoff = (lane & 15) * XPAD + ((lane >> 4) << 1);

    for (int t = 0; t < NTILE; ++t) {
        // wait for tile t's async chunks (in-order completion: <=4 outstanding
        // means the older group of 4 retired); final tile drains fully.
        if (t == NTILE - 1)
            asm volatile("s_wait_asynccnt 0x0" ::: "memory");
        else
            asm volatile("s_wait_asynccnt 0x4" ::: "memory");
        __syncthreads();                       // all waves' chunks visible

        const float* eb = es + (t & 1) * EBUF + eoff;

        v8f acc = {};
        #pragma unroll
        for (int ks = 0; ks < DD / 4; ++ks) {
            v2f b = *(const v2f*)(eb + ks * 4);
            acc = __builtin_amdgcn_wmma_f32_16x16x4_f32(
                false, areg[ks], false, b, (short)0, acc, false, false);
        }

        // fold into running argmin; C layout: lane holds N=lane&15,
        // rows M = i (lanes 0-15) or M = 8+i (lanes 16-31) in acc[i]
        float en = enorm[t * 16 + (lane & 15)];
        int kidx = t * 16 + (lane & 15);
        #pragma unroll
        for (int i = 0; i < 8; ++i) {
            float s = en - 2.0f * acc[i];
            if (s < rmin[i]) { rmin[i] = s; ridx[i] = kidx; }
        }

        __syncthreads();                       // everyone done reading buf[t&1]
        if (t + 2 < NTILE) prefetch(t + 2, t & 1);
    }

    // cross-lane argmin butterfly within each 16-lane half (tie -> lower index)
    #pragma unroll
    for (int i = 0; i < 8; ++i) {
        #pragma unroll
        for (int m = 1; m < 16; m <<= 1) {
            float om = __shfl_xor(rmin[i], m, 32);
            int   oi = __shfl_xor(ridx[i], m, 32);
            if (om < rmin[i] || (om == rmin[i] && oi < ridx[i])) {
                rmin[i] = om; ridx[i] = oi;
            }
        }
    }
    // lane 0 holds rows M=0..7, lane 16 holds rows M=8..15
    if ((lane & 15) == 0) {
        int base = rowBase + wave * 16 + ((lane >> 4) << 3);
        #pragma unroll
        for (int i = 0; i < 8; ++i) nmin[base + i] = ridx[i];
    }
}

// ---------------------------------------------------------------------------
// Kernel 4: gather z_q, weighted SOM scatter (L2 fp32 atomics), losses
//   one wave per x row; float4 for pure loads/stores, scalar f32 atomics
// ---------------------------------------------------------------------------
__global__ __launch_bounds__(256) void som_scatter(const float* __restrict__ X,
                                                   const float* __restrict__ E,
                                                   const int* __restrict__ nmin,
                                                   float* __restrict__ zq,
                                                   float* __restrict__ delta,
                                                   float* __restrict__ scalars) {
    int gw = (blockIdx.x * blockDim.x + threadIdx.x) >> 5;   // row index b
    int lane = threadIdx.x & 31;
    if (gw >= BB) return;

    int n = nmin[gw];
    int nx = n >> 5, ny = n & (SOMW - 1);
    int xu = (nx < SOMW - 1) ? nx + 1 : nx;
    int xd = (nx > 0)        ? nx - 1 : nx;
    int yr = (ny < SOMW - 1) ? ny + 1 : ny;
    int yl = (ny > 0)        ? ny - 1 : ny;
    int iu  = (xu << 5) | ny;
    int id_ = (xd << 5) | ny;
    int ir  = (nx << 5) | yr;
    int il  = (nx << 5) | yl;

    const float4* xr4 = (const float4*)(X + (size_t)gw * DD);
    const float4* er4 = (const float4*)(E + (size_t)n * DD);
    float4* zq4 = (float4*)(zq + (size_t)gw * DD);
    float commit = 0.0f, som = 0.0f;

    #pragma unroll
    for (int j = 0; j < 2; ++j) {
        int c4 = lane + j * 32;               // float4 column, 0..63
        float4 xv = xr4[c4];
        float4 ev = er4[c4];
        zq4[c4] = ev;

        float4 u  = ((const float4*)(E + (size_t)iu  * DD))[c4];
        float4 dn = ((const float4*)(E + (size_t)id_ * DD))[c4];
        float4 le = ((const float4*)(E + (size_t)il  * DD))[c4];
        float4 ri = ((const float4*)(E + (size_t)ir  * DD))[c4];

        float xs[4] = {xv.x, xv.y, xv.z, xv.w};
        float esv[4] = {ev.x, ev.y, ev.z, ev.w};
        float us[4] = {u.x, u.y, u.z, u.w};
        float ds_[4] = {dn.x, dn.y, dn.z, dn.w};
        float ls[4] = {le.x, le.y, le.z, le.w};
        float rs[4] = {ri.x, ri.y, ri.z, ri.w};

        #pragma unroll
        for (int q = 0; q < 4; ++q) {
            int d = c4 * 4 + q;
            float dw = xs[q] - esv[q];
            commit += dw * dw;
            atomicAdd(&delta[(size_t)n   * DD + d], LRATE * dw);
            atomicAdd(&delta[(size_t)iu  * DD + d], 0.5f * LRATE * dw);
            atomicAdd(&delta[(size_t)id_ * DD + d], 0.5f * LRATE * dw);
            atomicAdd(&delta[(size_t)ir  * DD + d], 0.5f * LRATE * dw);
            atomicAdd(&delta[(size_t)il  * DD + d], 0.5f * LRATE * dw);
            float e1 = us[q]  - xs[q];
            float e2 = ds_[q] - xs[q];
            float e3 = ls[q]  - xs[q];
            float e4 = rs[q]  - xs[q];
            som += e1 * e1 + e2 * e2 + e3 * e3 + e4 * e4;
        }
    }
    #pragma unroll
    for (int m = 16; m >= 1; m >>= 1) {
        commit += __shfl_xor(commit, m, 32);
        som    += __shfl_xor(som, m, 32);
    }
    if (lane == 0) {
        atomicAdd(&scalars[0], commit * (1.0f / ((float)BB * (float)DD)));
        atomicAdd(&scalars[1], som    * (1.0f / ((float)BB * 4.0f * (float)DD)));
    }
}

// ---------------------------------------------------------------------------
// Kernel 5: new_embeddings = embeddings + delta
// ---------------------------------------------------------------------------
__global__ __launch_bounds__(256) void som_finalize(const float* __restrict__ E,
                                                    const float* __restrict__ delta,
                                                    float* __restrict__ newe) {
    int i = blockIdx.x * blockDim.x + threadIdx.x;
    if (i < KK * DD) newe[i] = E[i] + delta[i];
}

// ---------------------------------------------------------------------------
extern "C" void kernel_launch(void* const* d_in, const int* in_sizes, int n_in,
                              void* d_out, int out_size, void* d_ws, size_t ws_size,
                              hipStream_t stream) {
    const float* X = (const float*)d_in[0];       // [B, D]
    const float* E = (const float*)d_in[1];       // [K, D]

    // output layout: z_q [B*D] | commit | som | new_embeddings [K*D]
    float* out     = (float*)d_out;
    float* zq      = out;
    float* scalars = out + (size_t)BB * DD;
    float* newe    = out + (size_t)BB * DD + 2;

    // workspace layout
    char* ws = (char*)d_ws;
    float* delta = (float*)ws;                                  // K*D f32 = 1 MB
    float* enorm = (float*)(ws + (size_t)KK * DD * 4);          // K f32
    int*   nmin  = (int*)  (ws + (size_t)KK * DD * 4 + KK * 4); // B i32

    // 1) zero accumulators
    som_init<<<(KK * DD) / 256, 256, 0, stream>>>(delta, scalars);

    // 2) embedding norms
    som_enorm<<<(KK * 32) / 256, 256, 0, stream>>>(E, enorm);

    // 3) fused distance + argmin (WMMA fp32, async double-buffered E tiles)
    som_argmin<<<BB / TILE_ROWS, 256, 0, stream>>>(X, E, enorm, nmin);

    // 4) gather + scatter + losses
    som_scatter<<<(BB * 32) / 256, 256, 0, stream>>>(X, E, nmin, zq, delta, scalars);

    // 5) new embeddings
    som_finalize<<<(KK * DD) / 256, 256, 0, stream>>>(E, delta, newe);
}